// ImgGCN_38482906972431
// MI455X (gfx1250) — compile-verified
//
#include <hip/hip_runtime.h>

typedef __attribute__((ext_vector_type(2))) float v2f;
typedef __attribute__((ext_vector_type(8))) float v8f;

#define NN 50000
#define EE 800000
#define DD 128
#define RR 128
#define OO 128
#define KCAT 384   // D + R + D
#define AGGW 256   // D + R

#define WMMA_F32(a, b, c) \
  __builtin_amdgcn_wmma_f32_16x16x4_f32(false, (a), false, (b), (short)0, (c), false, false)

// ---------------------------------------------------------------------------
// Kernel 1: h = node_feat @ Wn^T + bn      (one wave per 16x16 tile, K=128)
// ---------------------------------------------------------------------------
__global__ __launch_bounds__(256) void imggcn_node_fc(
    const float* __restrict__ X, const float* __restrict__ Wn,
    const float* __restrict__ bn, float* __restrict__ h) {
  int wave = (blockIdx.x * blockDim.x + threadIdx.x) >> 5;
  int lane = threadIdx.x & 31;
  int tile_n = wave & 7;        // 8 column tiles (128/16)
  int tile_m = wave >> 3;       // 3125 row tiles (50000/16)
  if (tile_m >= NN / 16) return;

  int mn = lane & 15;           // A row / B col handled by this lane
  int kh = (lane >> 4) * 2;     // K sub-offset for upper half-wave
  const float* arow = X  + (size_t)(tile_m * 16 + mn) * DD;
  const float* brow = Wn + (size_t)(tile_n * 16 + mn) * DD;  // B[k][n] = Wn[n][k]

  v8f acc = {};
#pragma unroll
  for (int k0 = 0; k0 < DD; k0 += 4) {
    v2f a = *(const v2f*)(arow + k0 + kh);
    v2f b = *(const v2f*)(brow + k0 + kh);
    acc = WMMA_F32(a, b, acc);
  }

  int col = tile_n * 16 + mn;
  float bias = bn[col];
  int rbase = (lane >> 4) * 8;
#pragma unroll
  for (int r = 0; r < 8; ++r) {
    int row = tile_m * 16 + rbase + r;
    h[(size_t)row * DD + col] = acc[r] + bias;
  }
}

// ---------------------------------------------------------------------------
// Kernel 2: z2 = edge_att * (edge_feat @ Wr^T + br), atomic scatter into
//           agg[dst, 128:256].
// Block = 256 edge-rows x 64 cols. B slab (64x128 = 32 KB) staged in LDS and
// shared by all 8 waves; each wave computes a 32x64 macro-tile with
// 2x4 register blocking (8 accumulators): per k-step 2 vmem + 4 ds loads
// feed 8 WMMAs.
// ---------------------------------------------------------------------------
__global__ __launch_bounds__(256) void imggcn_edge_fc_scatter(
    const float* __restrict__ Ef, const float* __restrict__ Wr,
    const float* __restrict__ br, const float* __restrict__ eatt,
    const int* __restrict__ dst, float* __restrict__ agg) {
  __shared__ float Bs[64 * RR];          // 32 KB: Wr rows [colmac*64, +64)

  int tid  = threadIdx.x;
  int wave = tid >> 5;
  int lane = tid & 31;
  int colmac = blockIdx.x & 1;           // which 64-col half of Wr
  int rowblk = blockIdx.x >> 1;          // 0..3124  (800000/256)

  // Cooperative stage of B: 8192 floats, 256 threads * 8 float4 each.
  {
    const float4* wsrc = (const float4*)(Wr + (size_t)colmac * 64 * RR);
    float4* bdst = (float4*)Bs;
#pragma unroll
    for (int i = 0; i < 8; ++i)
      bdst[tid + 256 * i] = wsrc[tid + 256 * i];
  }
  __syncthreads();

  int mn = lane & 15;
  int kh = (lane >> 4) * 2;
  int row0 = rowblk * 256 + wave * 32;   // this wave's 32 edge rows
  const float* arow0 = Ef + (size_t)(row0 + mn) * RR;
  const float* arow1 = arow0 + (size_t)16 * RR;
  const float* bs = Bs + (size_t)mn * RR;   // col-tile c adds c*16*RR

  v8f acc[2][4] = {};
#pragma unroll
  for (int k0 = 0; k0 < RR; k0 += 4) {
    v2f a0 = *(const v2f*)(arow0 + k0 + kh);
    v2f a1 = *(const v2f*)(arow1 + k0 + kh);
#pragma unroll
    for (int c = 0; c < 4; ++c) {
      v2f b = *(const v2f*)(bs + (size_t)c * 16 * RR + k0 + kh);
      acc[0][c] = WMMA_F32(a0, b, acc[0][c]);
      acc[1][c] = WMMA_F32(a1, b, acc[1][c]);
    }
  }

  int rbase = (lane >> 4) * 8;
#pragma unroll
  for (int rt = 0; rt < 2; ++rt) {
#pragma unroll
    for (int r = 0; r < 8; ++r) {
      int e = row0 + rt * 16 + rbase + r;
      float att = eatt[e];
      float* aggrow = agg + (size_t)dst[e] * AGGW + DD;
#pragma unroll
      for (int c = 0; c < 4; ++c) {
        int col = colmac * 64 + c * 16 + mn;
        float v = att * (acc[rt][c][r] + br[col]);
        atomicAdd(aggrow + col, v);      // L2-resident scatter
      }
    }
  }
}

// ---------------------------------------------------------------------------
// Kernel 3: z1 = node_att[src] * h[src] -> atomic scatter to agg[dst, 0:128];
//           deg counting folded in (one atomic per edge)
// ---------------------------------------------------------------------------
__global__ __launch_bounds__(256) void imggcn_z1_scatter(
    const float* __restrict__ h, const float* __restrict__ natt,
    const int* __restrict__ src, const int* __restrict__ dst,
    float* __restrict__ agg, float* __restrict__ deg) {
  int t = blockIdx.x * blockDim.x + threadIdx.x;
  if (t >= EE * (DD / 4)) return;       // 32 four-float chunks per edge
  int e  = t >> 5;
  int c4 = (t & 31) * 4;
  int s = src[e], d = dst[e];
  float a = natt[s];
  float4 hv = *(const float4*)(h + (size_t)s * DD + c4);
  float* ap = agg + (size_t)d * AGGW + c4;
  atomicAdd(ap + 0, a * hv.x);
  atomicAdd(ap + 1, a * hv.y);
  atomicAdd(ap + 2, a * hv.z);
  atomicAdd(ap + 3, a * hv.w);
  if ((t & 31) == 0) atomicAdd(deg + d, 1.0f);
}

// ---------------------------------------------------------------------------
// Kernel 4: out = where(deg>0, natt * relu([agg|h] @ Wa^T + ba), h)
//           K = 384 split into agg part (256) and h part (128)
// ---------------------------------------------------------------------------
__global__ __launch_bounds__(256) void imggcn_out_fc(
    const float* __restrict__ agg, const float* __restrict__ h,
    const float* __restrict__ Wa, const float* __restrict__ ba,
    const float* __restrict__ natt, const float* __restrict__ deg,
    float* __restrict__ out) {
  int wave = (blockIdx.x * blockDim.x + threadIdx.x) >> 5;
  int lane = threadIdx.x & 31;
  int tile_n = wave & 7;
  int tile_m = wave >> 3;
  if (tile_m >= NN / 16) return;

  int mn = lane & 15;
  int kh = (lane >> 4) * 2;
  int arow = tile_m * 16 + mn;
  const float* aggrow = agg + (size_t)arow * AGGW;
  const float* hrow   = h   + (size_t)arow * DD;
  const float* brow   = Wa  + (size_t)(tile_n * 16 + mn) * KCAT;

  v8f acc = {};
#pragma unroll
  for (int k0 = 0; k0 < AGGW; k0 += 4) {          // cat[:, 0:256] = agg
    v2f a = *(const v2f*)(aggrow + k0 + kh);
    v2f b = *(const v2f*)(brow + k0 + kh);
    acc = WMMA_F32(a, b, acc);
  }
#pragma unroll
  for (int k0 = 0; k0 < DD; k0 += 4) {            // cat[:, 256:384] = h
    v2f a = *(const v2f*)(hrow + k0 + kh);
    v2f b = *(const v2f*)(brow + AGGW + k0 + kh);
    acc = WMMA_F32(a, b, acc);
  }

  int col = tile_n * 16 + mn;
  float bias = ba[col];
  int rbase = (lane >> 4) * 8;
#pragma unroll
  for (int r = 0; r < 8; ++r) {
    int row = tile_m * 16 + rbase + r;
    float v = acc[r] + bias;
    v = v > 0.0f ? v : 0.0f;
    v *= natt[row];
    float hb = h[(size_t)row * DD + col];          // zero-in-degree fallback
    out[(size_t)row * OO + col] = (deg[row] > 0.0f) ? v : hb;
  }
}

// ---------------------------------------------------------------------------
extern "C" void kernel_launch(void* const* d_in, const int* in_sizes, int n_in,
                              void* d_out, int out_size, void* d_ws, size_t ws_size,
                              hipStream_t stream) {
  const float* node_feat = (const float*)d_in[0];
  const float* node_att  = (const float*)d_in[1];
  const float* edge_feat = (const float*)d_in[2];
  const float* edge_att  = (const float*)d_in[3];
  const float* Wn        = (const float*)d_in[4];
  const float* bn        = (const float*)d_in[5];
  const float* Wr        = (const float*)d_in[6];
  const float* br        = (const float*)d_in[7];
  const float* Wa        = (const float*)d_in[8];
  const float* ba        = (const float*)d_in[9];
  const int*   src       = (const int*)d_in[10];
  const int*   dst       = (const int*)d_in[11];
  float* out = (float*)d_out;

  // Workspace layout: h [N*D] | agg [N*256] | deg [N]   (~77 MB total)
  char* ws = (char*)d_ws;
  float* h   = (float*)ws;
  float* agg = (float*)(ws + (size_t)NN * DD * sizeof(float));
  float* deg = (float*)(ws + (size_t)NN * DD * sizeof(float)
                           + (size_t)NN * AGGW * sizeof(float));

  hipMemsetAsync(agg, 0, (size_t)NN * AGGW * sizeof(float), stream);
  hipMemsetAsync(deg, 0, (size_t)NN * sizeof(float), stream);

  int node_tiles = (NN / 16) * (DD / 16);   // 25000 (8 waves/block)
  imggcn_node_fc<<<node_tiles / 8, 256, 0, stream>>>(node_feat, Wn, bn, h);

  int edge_blocks = (EE / 256) * 2;         // 6250 macro-tile blocks
  imggcn_edge_fc_scatter<<<edge_blocks, 256, 0, stream>>>(
      edge_feat, Wr, br, edge_att, dst, agg);

  int z1_threads = EE * (DD / 4);           // 25.6M
  imggcn_z1_scatter<<<(z1_threads + 255) / 256, 256, 0, stream>>>(
      h, node_att, src, dst, agg, deg);

  imggcn_out_fc<<<node_tiles / 8, 256, 0, stream>>>(
      agg, h, Wa, ba, node_att, deg, out);
}